// Hete_MLP2_atten_69836168233642
// MI455X (gfx1250) — compile-verified
//
#include <hip/hip_runtime.h>
#include <hip/hip_bf16.h>
#include <math.h>

typedef __attribute__((ext_vector_type(2))) float v2f;
typedef __attribute__((ext_vector_type(8))) float v8f;

#define META 8
#define DIM 256
#define HID 128
#define TN 16           // nodes per workgroup tile
#define XPITCH 260      // 256 + 4 pad: perfect bank spread for b64 A-frag loads, 16B aligned rows
#define WPITCH 136      // 128 + 8 pad: disjoint bank sets for the two half-wave B rows, 16B aligned rows

// Branch-free tanh: prefer the CDNA5 hardware v_tanh_f32 transcendental; otherwise
// a no-branch rational form (exp is v_exp_f32, rcp is v_rcp_f32 -> no exec-mask churn).
__device__ __forceinline__ float fast_tanhf(float v) {
#if __has_builtin(__builtin_amdgcn_tanhf)
    return __builtin_amdgcn_tanhf(v);
#elif __has_builtin(__builtin_amdgcn_tanh_f32)
    return __builtin_amdgcn_tanh_f32(v);
#else
    const float a = __builtin_fabsf(v);
    const float e = __expf(-2.0f * a);          // in (0,1], no overflow for any a
    const float r = (1.0f - e) * __builtin_amdgcn_rcpf(1.0f + e);
    return __builtin_copysignf(r, v);
#endif
}

__global__ __launch_bounds__(256)
void hete_mlp2_atten_kernel(const float* __restrict__ x,   // [M,N,D]
                            const float* __restrict__ W1,  // [D,H] row-major (d major)
                            const float* __restrict__ b1,  // [H]
                            const float* __restrict__ W2,  // [H]
                            const float* __restrict__ b2,  // [1]
                            float* __restrict__ out,       // [N,D]
                            int N)
{
    __shared__ __align__(16) float xs[META][TN][XPITCH];   // 133,120 B
    __shared__ __align__(16) float w1s[DIM][WPITCH];       // 139,264 B
    __shared__ float lgs[META][TN];                        // logits
    __shared__ float atts[META][TN];                       // softmax weights

    const int t  = threadIdx.x;
    const int n0 = blockIdx.x * TN;

    // ---------- Phase 0: stage x tile [8][16][256] into LDS (float4, coalesced) ----------
    for (int f = t; f < META * TN * (DIM / 4); f += 256) {
        const int m  = f >> 10;        // 16*64 float4 per meta path
        const int r  = f & 1023;
        const int nl = r >> 6;         // node within tile
        const int d4 = r & 63;         // float4 index along D
        float4 v;
        if (n0 + nl < N) {
            v = *(const float4*)(x + ((size_t)m * N + n0 + nl) * DIM + (size_t)d4 * 4);
        } else {
            v = make_float4(0.f, 0.f, 0.f, 0.f);
        }
        *(float4*)&xs[m][nl][d4 * 4] = v;
    }
    // ---------- stage W1 [256][128] into LDS ----------
    for (int f = t; f < DIM * (HID / 4); f += 256) {
        const int k  = f >> 5;
        const int d4 = f & 31;
        *(float4*)&w1s[k][d4 * 4] = *(const float4*)(W1 + (size_t)k * HID + (size_t)d4 * 4);
    }
    __syncthreads();

    // ---------- Phase 1: per-wave fp32 WMMA GEMM h = tanh(x@W1 + b1), logits = h . W2 ----------
    const int lane = t & 31;
    const int m    = t >> 5;       // wave id == meta path (8 waves)
    const int lo   = lane & 15;
    const int hi   = lane >> 4;    // half-wave select
    const int koff = hi * 2;       // K offset of this half (ISA 16x4 A / 4x16 B layout)

    float b1v[8], w2v[8];
    #pragma unroll
    for (int ct = 0; ct < 8; ++ct) {
        const int col = ct * 16 + lo;
        b1v[ct] = b1[col];
        w2v[ct] = W2[col];
    }

    v8f acc[8];                     // 8 column tiles of the 16x128 hidden block -> 64 VGPRs
    #pragma unroll
    for (int ct = 0; ct < 8; ++ct)
        #pragma unroll
        for (int r = 0; r < 8; ++r) acc[ct][r] = 0.0f;

    for (int kb = 0; kb < DIM; kb += 4) {
        // A-frag: rows = 16 nodes, K = kb..kb+3 split across halves (ds_load_b64, bank-perfect)
        v2f a = *(const v2f*)&xs[m][lo][kb + koff];
        #pragma unroll
        for (int ct = 0; ct < 8; ++ct) {
            v2f b;
            b[0] = w1s[kb + koff + 0][ct * 16 + lo];
            b[1] = w1s[kb + koff + 1][ct * 16 + lo];
            acc[ct] = __builtin_amdgcn_wmma_f32_16x16x4_f32(
                false, a, false, b, (short)0, acc[ct], false, false);
        }
    }

    // tanh + dot with W2: per-lane column partials, then 16-lane reduction per half-wave
    float lp[8];
    #pragma unroll
    for (int r = 0; r < 8; ++r) lp[r] = 0.0f;
    #pragma unroll
    for (int ct = 0; ct < 8; ++ct) {
        const float b1c = b1v[ct], w2c = w2v[ct];
        #pragma unroll
        for (int r = 0; r < 8; ++r)
            lp[r] += fast_tanhf(acc[ct][r] + b1c) * w2c;
    }
    #pragma unroll
    for (int r = 0; r < 8; ++r) {
        float v = lp[r];
        v += __shfl_xor(v, 1, 16);
        v += __shfl_xor(v, 2, 16);
        v += __shfl_xor(v, 4, 16);
        v += __shfl_xor(v, 8, 16);
        if (lo == 0) lgs[m][hi * 8 + r] = v;   // C layout: VGPR r -> row r (half0) / r+8 (half1)
    }
    __syncthreads();

    // ---------- Phase 2: softmax over meta paths (per node) ----------
    if (t < TN) {
        const float bb = b2[0];
        float l[META];
        float mx = -1e30f;
        #pragma unroll
        for (int mm = 0; mm < META; ++mm) {
            l[mm] = lgs[mm][t] + bb;
            mx = fmaxf(mx, l[mm]);
        }
        float s = 0.f;
        #pragma unroll
        for (int mm = 0; mm < META; ++mm) {
            l[mm] = __expf(l[mm] - mx);
            s += l[mm];
        }
        const float inv = 1.0f / s;
        #pragma unroll
        for (int mm = 0; mm < META; ++mm) atts[mm][t] = l[mm] * inv;
    }
    __syncthreads();

    // ---------- Phase 3: out[n,d] = sum_m att[m,n] * x[m,n,d]  (thread = d, coalesced stores) ----------
    #pragma unroll 1
    for (int i = 0; i < TN; ++i) {
        if (n0 + i >= N) break;
        float s = 0.f;
        #pragma unroll
        for (int mm = 0; mm < META; ++mm)
            s += atts[mm][i] * xs[mm][i][t];   // att broadcast read; xs conflict-free (lane==bank run)
        out[(size_t)(n0 + i) * DIM + t] = s;
    }
}

extern "C" void kernel_launch(void* const* d_in, const int* in_sizes, int n_in,
                              void* d_out, int out_size, void* d_ws, size_t ws_size,
                              hipStream_t stream) {
    // inputs: [0] nodes(int32, unused by reference math), [1] homo_embeddings f32 [M,N,D],
    //         [2] W1 f32 [D,H], [3] b1 f32 [H], [4] W2 f32 [H], [5] b2 f32 scalar
    const float* x  = (const float*)d_in[1];
    const float* W1 = (const float*)d_in[2];
    const float* b1 = (const float*)d_in[3];
    const float* W2 = (const float*)d_in[4];
    const float* b2 = (const float*)d_in[5];
    float* out = (float*)d_out;

    const int N = in_sizes[0];                 // node count (100000)
    const int grid = (N + TN - 1) / TN;        // 6250 workgroups of 256 threads (8 waves)
    hete_mlp2_atten_kernel<<<grid, 256, 0, stream>>>(x, W1, b1, W2, b2, out, N);
}